// diagonal_solver_833223655861
// MI455X (gfx1250) — compile-verified
//
#include <hip/hip_runtime.h>
#include <cstdint>
#include <cstddef>

// SIZE = 4^6 = 4096; x is 4096x4096 fp32 (64 MiB). Memory-bound diagonal
// similarity transform: out[i][j] = m[i] * x[i][j] * (1/m)[j].
static constexpr int N = 4096;
static constexpr int THREADS = 256;          // 8 wave32s per block
static constexpr int ROWS_PER_BLOCK = 4;     // amortize the LDS rinv stage
static constexpr int VEC_PER_ROW = N / 4;    // 1024 float4 per row
static constexpr int STAGE_ITERS = VEC_PER_ROW / THREADS; // 4

// Pointer types matching the async-copy builtin's signature:
//   void __builtin_amdgcn_global_load_async_to_lds_b128(v4i AS1*, v4i AS3*, Ii, Ii)
typedef int v4i __attribute__((ext_vector_type(4)));
typedef __attribute__((address_space(1))) v4i gv4i;   // global (AS1)
typedef __attribute__((address_space(3))) v4i lv4i;   // LDS (AS3)

// ---------------------------------------------------------------------------
// Kernel 1: m[i] = p[d0]*p[d1]*...*p[d5] (base-4 digits of i, left-to-right
// fold to match the reference's repeated vector-kron), rinv[i] = 1/m[i].
// ---------------------------------------------------------------------------
__global__ __launch_bounds__(THREADS)
void kron_scales_kernel(const float* __restrict__ p,
                        float* __restrict__ m,
                        float* __restrict__ rinv) {
    const int i = blockIdx.x * blockDim.x + threadIdx.x;
    if (i >= N) return;
    float v = p[(i >> 10) & 3];
    v *= p[(i >> 8) & 3];
    v *= p[(i >> 6) & 3];
    v *= p[(i >> 4) & 3];
    v *= p[(i >> 2) & 3];
    v *= p[i & 3];
    m[i] = v;
    rinv[i] = 1.0f / v;
}

// ---------------------------------------------------------------------------
// Kernel 2: streaming diagonal scale. One block handles 4 rows (64 KiB of x).
// rinv (16 KiB) is staged into LDS via the CDNA5 async global->LDS copy
// (ASYNCcnt path), then each thread streams float4s: 128-bit coalesced
// loads/stores, row scale m[row] is wave-uniform (scalar load).
// ---------------------------------------------------------------------------
__global__ __launch_bounds__(THREADS)
void diag_scale_kernel(const float4* __restrict__ x,
                       const float* __restrict__ m,
                       const float4* __restrict__ rinv,
                       float4* __restrict__ out) {
    __shared__ float4 lds_rinv[VEC_PER_ROW];  // 16 KiB
    const int tid = threadIdx.x;
    const int row0 = blockIdx.x * ROWS_PER_BLOCK;

#if defined(__gfx1250__) && \
    __has_builtin(__builtin_amdgcn_global_load_async_to_lds_b128) && \
    __has_builtin(__builtin_amdgcn_s_wait_asynccnt)
    // CDNA5 async staging: per-lane 16B global->LDS DMA, tracked with
    // ASYNCcnt (no VGPR round-trip).
#pragma unroll
    for (int k = 0; k < STAGE_ITERS; ++k) {
        const int idx = k * THREADS + tid;
        __builtin_amdgcn_global_load_async_to_lds_b128(
            (gv4i*)(v4i*)(rinv + idx),
            (lv4i*)(v4i*)(&lds_rinv[idx]),
            /*offset=*/0, /*cpol=*/0);
    }
    __builtin_amdgcn_s_wait_asynccnt(0);   // s_wait_asynccnt 0
#else
#pragma unroll
    for (int k = 0; k < STAGE_ITERS; ++k) {
        const int idx = k * THREADS + tid;
        lds_rinv[idx] = rinv[idx];
    }
#endif
    __syncthreads();

#pragma unroll
    for (int r = 0; r < ROWS_PER_BLOCK; ++r) {
        const int row = row0 + r;
        const float mr = m[row];                           // wave-uniform
        const float4* __restrict__ xrow = x + (size_t)row * VEC_PER_ROW;
        float4* __restrict__ orow = out + (size_t)row * VEC_PER_ROW;
#pragma unroll
        for (int k = 0; k < STAGE_ITERS; ++k) {
            const int idx = k * THREADS + tid;
            const float4 xv = xrow[idx];                   // global_load_b128
            const float4 rv = lds_rinv[idx];               // ds_load_b128
            float4 ov;
            ov.x = (mr * xv.x) * rv.x;
            ov.y = (mr * xv.y) * rv.y;
            ov.z = (mr * xv.z) * rv.z;
            ov.w = (mr * xv.w) * rv.w;
            orow[idx] = ov;                                // global_store_b128
        }
    }
}

// ---------------------------------------------------------------------------
// Harness entry point.
//   d_in[0] = x       (4096*4096 fp32)
//   d_in[1] = params3 (4 fp32)
//   d_out   = out     (4096*4096 fp32)
//   d_ws    : [0, N)      -> m
//             [N, 2N)     -> rinv
// ---------------------------------------------------------------------------
extern "C" void kernel_launch(void* const* d_in, const int* in_sizes, int n_in,
                              void* d_out, int out_size, void* d_ws, size_t ws_size,
                              hipStream_t stream) {
    (void)in_sizes; (void)n_in; (void)out_size; (void)ws_size;
    const float* x = (const float*)d_in[0];
    const float* p = (const float*)d_in[1];
    float* out  = (float*)d_out;
    float* m    = (float*)d_ws;
    float* rinv = m + N;

    kron_scales_kernel<<<N / THREADS, THREADS, 0, stream>>>(p, m, rinv);

    diag_scale_kernel<<<N / ROWS_PER_BLOCK, THREADS, 0, stream>>>(
        (const float4*)x, m, (const float4*)rinv, (float4*)out);
}